// QueryAndGroupNorm_49185965474467
// MI455X (gfx1250) — compile-verified
//
#include <hip/hip_runtime.h>
#include <stdint.h>
#include <math.h>

// Problem constants (match reference setup_inputs)
#define Bv 4
#define Nv 16384
#define Mv 2048
#define Sv 32          // SAMPLE_NUM
#define Cv 64          // CHANNEL
#define CHv 67         // 3 + CHANNEL
#define R2MAX 0.04f    // MAX_RADIUS^2
#define EPSd 1e-5

// Tiling
#define CPB 8                      // centers per block (1 wave per center)
#define TILE 4096                  // points staged in LDS per tile (48 KB)
#define NTILES (Nv / TILE)         // 4
#define BLKS_PER_B (Mv / CPB)      // 256
#define NBLK (Bv * BLKS_PER_B)     // 1024 blocks total in kernel 1

// ---------------------------------------------------------------------------
// Kernel 1: ball query (wave32 ballot ordered-append) + gather + centered
// values written in (B,CH,M,S) layout, plus per-block double partial sums.
// Point tiles are streamed into LDS with gfx1250 async-to-LDS loads.
// ---------------------------------------------------------------------------
__global__ __launch_bounds__(256) void k_group(
    const float* __restrict__ pts,    // (B,N,3)
    const float* __restrict__ cxyz,   // (B,M,3)
    const float* __restrict__ cfeat,  // (B,M,C)
    const float* __restrict__ feats,  // (B,C,N)
    float* __restrict__ out,          // (B,CH,M,S) centered values
    double* __restrict__ partials)    // [NBLK] sums, [NBLK] sumsq
{
  __shared__ __align__(16) float tile[TILE * 3];   // 48 KB staged points
  __shared__ int    sIdx[CPB][Sv];
  __shared__ double sred[CPB], qred[CPB];

  const int tid  = threadIdx.x;
  const int w    = tid >> 5;      // wave id within block = local center
  const int lane = tid & 31;
  const int b    = blockIdx.x / BLKS_PER_B;
  const int cb   = blockIdx.x % BLKS_PER_B;
  const int m    = cb * CPB + w;

  const float cx = cxyz[(b * Mv + m) * 3 + 0];
  const float cy = cxyz[(b * Mv + m) * 3 + 1];
  const float cz = cxyz[(b * Mv + m) * 3 + 2];

  const unsigned lds_base = (unsigned)(uintptr_t)(void*)&tile[0];
  int cnt = 0;   // uniform within wave

  for (int t = 0; t < NTILES; ++t) {
    // --- async stream one tile of points (TILE*3 floats) into LDS ---------
    const float* gsrc = pts + ((size_t)b * Nv + (size_t)t * TILE) * 3;
    #pragma unroll
    for (int i = 0; i < (TILE * 3 * 4) / (256 * 16); ++i) {   // 12 x b128/thread
      unsigned byteoff = (unsigned)(i * 256 + tid) * 16u;
      unsigned laddr   = lds_base + byteoff;
      asm volatile("global_load_async_to_lds_b128 %0, %1, %2"
                   :: "v"(laddr), "v"(byteoff), "s"(gsrc) : "memory");
    }
    asm volatile("s_wait_asynccnt 0" ::: "memory");
    __syncthreads();   // all waves' async copies complete -> tile valid

    // --- wave32 ordered scan: first Sv in-range indices -------------------
    if (cnt < Sv) {
      const int base = t * TILE;
      for (int p0 = 0; p0 < TILE; p0 += 32) {
        const int p = p0 + lane;
        const float dx = tile[p * 3 + 0] - cx;
        const float dy = tile[p * 3 + 1] - cy;
        const float dz = tile[p * 3 + 2] - cz;
        const float d2 = dx * dx + dy * dy + dz * dz;
        const bool  pred = d2 < R2MAX;                   // MIN_RADIUS==0
        const unsigned mask   = (unsigned)__ballot(pred);
        const int      prefix = __popc(mask & ((1u << lane) - 1u));
        if (pred && (cnt + prefix) < Sv) sIdx[w][cnt + prefix] = base + p;
        cnt += __popc(mask);
        if (cnt >= Sv) { cnt = Sv; break; }
      }
    }
    // barrier doubles as "tile may be overwritten" fence + early-out vote
    if (__syncthreads_and(cnt >= Sv)) break;
  }

  // fill remaining slots with first found index (0 if ball empty)
  {
    const int first = (cnt > 0) ? sIdx[w][0] : 0;
    const int v = (lane < cnt) ? sIdx[w][lane] : first;
    sIdx[w][lane] = v;
  }
  __syncthreads();

  // --- gather + center + write (B,CH,M,S), accumulate sum/sumsq -----------
  const int idx = sIdx[w][lane];
  float asum = 0.f, asq = 0.f;
  const float* prow  = pts + ((size_t)b * Nv + (size_t)idx) * 3;
  float* obase = out + (((size_t)b * CHv) * Mv + (size_t)m) * Sv + lane;

  #pragma unroll
  for (int c = 0; c < 3; ++c) {
    const float val = prow[c] - cxyz[(b * Mv + m) * 3 + c];
    obase[(size_t)c * (Mv * Sv)] = val;
    asum += val; asq += val * val;
  }
  #pragma unroll 4
  for (int c = 0; c < Cv; ++c) {
    const float val = feats[((size_t)b * Cv + c) * Nv + idx]
                    - cfeat[((size_t)b * Mv + m) * Cv + c];
    obase[(size_t)(3 + c) * (Mv * Sv)] = val;
    asum += val; asq += val * val;
  }

  // wave32 butterfly reduce, then block reduce (deterministic, no atomics)
  #pragma unroll
  for (int o = 16; o > 0; o >>= 1) {
    asum += __shfl_xor(asum, o);
    asq  += __shfl_xor(asq, o);
  }
  if (lane == 0) { sred[w] = (double)asum; qred[w] = (double)asq; }
  __syncthreads();
  if (tid == 0) {
    double s = 0.0, q = 0.0;
    #pragma unroll
    for (int i = 0; i < CPB; ++i) { s += sred[i]; q += qred[i]; }
    partials[blockIdx.x]        = s;
    partials[NBLK + blockIdx.x] = q;
  }
}

// ---------------------------------------------------------------------------
// Kernel 2: per-batch reduction of 256 partials -> scale = 1/(std+eps)
// ---------------------------------------------------------------------------
__global__ __launch_bounds__(256) void k_reduce(
    const double* __restrict__ partials, float* __restrict__ scales)
{
  __shared__ double rs[256], rq[256];
  const int b = blockIdx.x, tid = threadIdx.x;
  rs[tid] = partials[b * BLKS_PER_B + tid];
  rq[tid] = partials[NBLK + b * BLKS_PER_B + tid];
  __syncthreads();
  for (int o = 128; o > 0; o >>= 1) {
    if (tid < o) { rs[tid] += rs[tid + o]; rq[tid] += rq[tid + o]; }
    __syncthreads();
  }
  if (tid == 0) {
    const double n   = (double)CHv * Mv * Sv;            // 4390912
    const double var = (rq[0] - rs[0] * rs[0] / n) / (n - 1.0);
    const double sd  = sqrt(var);
    scales[b] = (float)(1.0 / (sd + EPSd));
  }
}

// ---------------------------------------------------------------------------
// Kernel 3: in-place float4 affine: out = alpha[c]*val*scale[b] + beta[c]
// ---------------------------------------------------------------------------
__global__ __launch_bounds__(256) void k_scale(
    float* __restrict__ out,
    const float* __restrict__ alpha, const float* __restrict__ beta,
    const float* __restrict__ scales)
{
  const size_t n4 = (size_t)Bv * CHv * Mv * Sv / 4;
  const size_t i  = (size_t)blockIdx.x * blockDim.x + threadIdx.x;
  if (i >= n4) return;
  const size_t e = i * 4;
  const int chunk = (int)(e >> 16);        // Mv*Sv = 65536 elems per (b,c)
  const int c = chunk % CHv;
  const int b = chunk / CHv;
  const float sc = scales[b];
  const float a  = alpha[c] * sc;
  const float bt = beta[c];
  float4 v = reinterpret_cast<float4*>(out)[i];
  v.x = v.x * a + bt;  v.y = v.y * a + bt;
  v.z = v.z * a + bt;  v.w = v.w * a + bt;
  reinterpret_cast<float4*>(out)[i] = v;
}

// ---------------------------------------------------------------------------
extern "C" void kernel_launch(void* const* d_in, const int* in_sizes, int n_in,
                              void* d_out, int out_size, void* d_ws, size_t ws_size,
                              hipStream_t stream) {
  const float* pts   = (const float*)d_in[0];   // (B,N,3)
  const float* cxyz  = (const float*)d_in[1];   // (B,M,3)
  const float* cfeat = (const float*)d_in[2];   // (B,M,C)
  const float* feats = (const float*)d_in[3];   // (B,C,N)
  const float* alpha = (const float*)d_in[4];   // (67)
  const float* beta  = (const float*)d_in[5];   // (67)
  float* out = (float*)d_out;

  double* partials = (double*)d_ws;                       // 2*NBLK doubles
  float*  scales   = (float*)(partials + 2 * NBLK);       // 4 floats

  k_group<<<NBLK, 256, 0, stream>>>(pts, cxyz, cfeat, feats, out, partials);
  k_reduce<<<Bv, 256, 0, stream>>>(partials, scales);

  const size_t n4 = (size_t)Bv * CHv * Mv * Sv / 4;       // 4,390,912
  const int grid3 = (int)((n4 + 255) / 256);
  k_scale<<<grid3, 256, 0, stream>>>(out, alpha, beta, scales);
}